// KoopmanLQR_65566970740964
// MI455X (gfx1250) — compile-verified
//
#include <hip/hip_runtime.h>

#define BSZ   2048
#define TT    30
#define LDP   66   // padded leading dim for 64x64 LDS arrays (breaks bank conflicts)

typedef __attribute__((ext_vector_type(2))) float v2f;
typedef __attribute__((ext_vector_type(8))) float v8f;

// ---- WMMA f32 16x16x4 fragment helpers (wave32 lane layouts per CDNA5 ISA) ----
// A (16x4): lane L holds row M=L&15, K = 2*(L>>4)+{0,1} in vgpr {0,1}
// B (4x16): lane L holds col N=L&15, K = 2*(L>>4)+{0,1} in vgpr {0,1}
// C/D (16x16): vgpr q, lane L -> M = q + 8*(L>>4), N = L&15

__device__ __forceinline__ v2f ldA(const float* S, int ld, int mr, int kc, int lane) {
  int m = mr + (lane & 15);
  int k = kc + 2 * (lane >> 4);
  v2f a; a.x = S[m * ld + k]; a.y = S[m * ld + k + 1]; return a;
}
// logical A[m][k] = sgn * S[k][m]  (transposed source)
__device__ __forceinline__ v2f ldAT(const float* S, int ld, int mr, int kc, int lane, float sgn) {
  int m = mr + (lane & 15);
  int k = kc + 2 * (lane >> 4);
  v2f a; a.x = sgn * S[k * ld + m]; a.y = sgn * S[(k + 1) * ld + m]; return a;
}
__device__ __forceinline__ v2f ldB(const float* S, int ld, int kc, int nc, int lane) {
  int n = nc + (lane & 15);
  int k = kc + 2 * (lane >> 4);
  v2f b; b.x = S[k * ld + n]; b.y = S[(k + 1) * ld + n]; return b;
}
__device__ __forceinline__ void stC(float* S, int ld, int mr, int nc, int lane, v8f c) {
  int n = nc + (lane & 15);
  int mb = mr + 8 * (lane >> 4);
#pragma unroll
  for (int q = 0; q < 8; ++q) S[(mb + q) * ld + n] = c[q];
}

#define WMMA(c, a, b) \
  __builtin_amdgcn_wmma_f32_16x16x4_f32(false, (a), false, (b), (short)0, (c), false, false)

__global__ void __launch_bounds__(256)
koopman_lqr_kernel(const float* __restrict__ gA, const float* __restrict__ gB,
                   const float* __restrict__ gQ, const float* __restrict__ gR,
                   const float* __restrict__ gGoals, float* __restrict__ out)
{
  __shared__ float sV [64 * LDP];   // running V (symmetric), padded
  __shared__ float sM2[64 * LDP];   // M2 = A^T V, padded
  __shared__ float sB [64 * 16];    // B
  __shared__ float sVB[64 * 16];    // V*B
  __shared__ float sG [16 * 64];    // G = B^T V A  ( = (VB)^T A, V symmetric)
  __shared__ float sKg[16 * 64];    // Kg = Vuu^-1 G
  __shared__ float sAug[16 * 32];   // [Vuu | I] -> [I | Vuu^-1]
  __shared__ float sRow[32];
  __shared__ float sv [64];
  __shared__ float sv2[64];
  __shared__ float sbv[16];
  __shared__ float skg[16];

  const int b    = blockIdx.x;
  const int tid  = threadIdx.x;
  const int lane = tid & 31;
  const int w    = tid >> 5;

  const float* Ab = gA + (size_t)b * 64 * 64;
  const float* Qb = gQ + (size_t)b * 64 * 64;
  const float* Bb = gB + (size_t)b * 64 * 16;
  const float* Rb = gR + (size_t)b * 16 * 16;
  const float* gb = gGoals + (size_t)b * (TT + 1) * 64;

  // init: V = Q, stage B; v0 = Q @ goals[:, T]
  for (int i = tid; i < 64 * 64; i += 256) sV[(i >> 6) * LDP + (i & 63)] = Qb[i];
  for (int i = tid; i < 64 * 16; i += 256) sB[i] = Bb[i];
  if (tid < 64) {
    const float* gT = gb + TT * 64;
    float acc = 0.f;
    for (int j = 0; j < 64; ++j) acc += Qb[tid * 64 + j] * gT[j];
    sv[tid] = acc;
  }
  __syncthreads();

  for (int step = TT - 1; step >= 0; --step) {
    // ---- Phase 1: VB = V * B   (64x16, 4 tiles, waves 0..3) ----
    if (w < 4) {
      const int mr = w * 16;
      v8f c = {};
      for (int k = 0; k < 64; k += 4) {
        v2f a  = ldA(sV, LDP, mr, k, lane);
        v2f bb = ldB(sB, 16, k, 0, lane);
        c = WMMA(c, a, bb);
      }
      stC(sVB, 16, mr, 0, lane, c);
    }
    __syncthreads();

    // ---- Phase 2: G = (VB)^T * A (waves 0..3) | Vuu = B^T*VB (wave 4) | bv = B^T v (wave 5) ----
    if (w < 4) {
      const int nc = w * 16;
      v8f c = {};
      for (int k = 0; k < 64; k += 4) {
        v2f a = ldAT(sVB, 16, 0, k, lane, 1.f);
        v2f bb;
        { int n = nc + (lane & 15); int kk = k + 2 * (lane >> 4);
          bb.x = Ab[kk * 64 + n]; bb.y = Ab[(kk + 1) * 64 + n]; }
        c = WMMA(c, a, bb);
      }
      stC(sG, 64, 0, nc, lane, c);
    } else if (w == 4) {
      v8f c = {};
      for (int k = 0; k < 64; k += 4) {
        v2f a  = ldAT(sB, 16, 0, k, lane, 1.f);   // B^T
        v2f bb = ldB(sVB, 16, k, 0, lane);
        c = WMMA(c, a, bb);
      }
      stC(sAug, 32, 0, 0, lane, c);               // left half of augmented matrix
    } else if (w == 5 && lane < 16) {
      float acc = 0.f;
      for (int j = 0; j < 64; ++j) acc += sB[j * 16 + lane] * sv[j];
      sbv[lane] = acc;
    }
    __syncthreads();

    // ---- Phase 3: M2 = A^T * V (64x64, 16 tiles, all 8 waves) + augmented setup ----
    for (int t = w * 2; t < w * 2 + 2; ++t) {
      const int mr = (t >> 2) * 16;
      const int nc = (t & 3) * 16;
      v8f c = {};
      for (int k = 0; k < 64; k += 4) {
        v2f a;
        { int m = mr + (lane & 15); int kk = k + 2 * (lane >> 4);
          a.x = Ab[kk * 64 + m]; a.y = Ab[(kk + 1) * 64 + m]; }  // A^T
        v2f bb = ldB(sV, LDP, k, nc, lane);
        c = WMMA(c, a, bb);
      }
      stC(sM2, LDP, mr, nc, lane, c);
    }
    for (int i = tid; i < 16 * 16; i += 256) {    // Vuu += R ; right half = I
      int r = i >> 4, cc = i & 15;
      sAug[r * 32 + cc] += Rb[i];
      sAug[r * 32 + 16 + cc] = (r == cc) ? 1.f : 0.f;
    }
    __syncthreads();

    // ---- Phase 4: Gauss-Jordan inversion of Vuu (SPD, no pivoting needed) ----
    for (int k = 0; k < 16; ++k) {
      if (tid < 32) sRow[tid] = sAug[k * 32 + tid];
      __syncthreads();
      if (tid < 16) {
        float piv = sRow[k];
        if (tid == k) {
          float inv = 1.f / piv;
          for (int j = 0; j < 32; ++j) sAug[k * 32 + j] = sRow[j] * inv;
        } else {
          float fac = sAug[tid * 32 + k] / piv;
          for (int j = 0; j < 32; ++j) sAug[tid * 32 + j] -= fac * sRow[j];
        }
      }
      __syncthreads();
    }

    // ---- Phase 5: Kg = Vuu^-1 * G (waves 0..3) | kg = Vuu^-1 * bv (wave 4) ----
    if (w < 4) {
      const int nc = w * 16;
      v8f c = {};
      for (int k = 0; k < 16; k += 4) {
        v2f a  = ldA(sAug + 16, 32, 0, k, lane);  // Vuu^-1 (right half, ld 32)
        v2f bb = ldB(sG, 64, k, nc, lane);
        c = WMMA(c, a, bb);
      }
      stC(sKg, 64, 0, nc, lane, c);
    } else if (w == 4 && lane < 16) {
      float acc = 0.f;
      for (int j = 0; j < 16; ++j) acc += sAug[lane * 32 + 16 + j] * sbv[j];
      skg[lane] = acc;
    }
    __syncthreads();

    // ---- Phase 6: V' = M2*A - G^T*Kg + Q   (16 tiles, all 8 waves) -> sV ----
    for (int t = w * 2; t < w * 2 + 2; ++t) {
      const int mr = (t >> 2) * 16;
      const int nc = (t & 3) * 16;
      v8f c;
      { int n = nc + (lane & 15); int mb = mr + 8 * (lane >> 4);
#pragma unroll
        for (int q = 0; q < 8; ++q) c[q] = Qb[(mb + q) * 64 + n]; }
      for (int k = 0; k < 64; k += 4) {
        v2f a = ldA(sM2, LDP, mr, k, lane);
        v2f bb;
        { int n = nc + (lane & 15); int kk = k + 2 * (lane >> 4);
          bb.x = Ab[kk * 64 + n]; bb.y = Ab[(kk + 1) * 64 + n]; }
        c = WMMA(c, a, bb);
      }
      for (int k = 0; k < 16; k += 4) {
        v2f a  = ldAT(sG, 64, mr, k, lane, -1.f);  // -(G^T) = -(A^T V B)
        v2f bb = ldB(sKg, 64, k, nc, lane);
        c = WMMA(c, a, bb);
      }
      stC(sV, LDP, mr, nc, lane, c);
    }
    __syncthreads();

    // ---- Phase 7: write [Kg | kg] for this step; v' = A^T v - Kg^T bv + Q*goal ----
    {
      float* outp = out + ((size_t)step * BSZ + b) * (16 * 65);
      for (int i = tid; i < 16 * 65; i += 256) {
        int u = i / 65, cc = i % 65;
        outp[i] = (cc < 64) ? sKg[u * 64 + cc] : skg[u];
      }
      if (tid < 64) {
        const float* gi = gb + step * 64;
        float acc = 0.f;
        for (int j = 0; j < 64; ++j) acc += Ab[j * 64 + tid] * sv[j];   // (A^T v)
        for (int j = 0; j < 16; ++j) acc -= sKg[j * 64 + tid] * sbv[j]; // (Kg^T bv)
        float accq = 0.f;
        for (int j = 0; j < 64; ++j) accq += Qb[tid * 64 + j] * gi[j];  // (Q g)
        sv2[tid] = acc + accq;
      }
    }
    __syncthreads();
    if (tid < 64) sv[tid] = sv2[tid];
    __syncthreads();
  }
}

extern "C" void kernel_launch(void* const* d_in, const int* in_sizes, int n_in,
                              void* d_out, int out_size, void* d_ws, size_t ws_size,
                              hipStream_t stream) {
  (void)in_sizes; (void)n_in; (void)out_size; (void)d_ws; (void)ws_size;
  const float* A     = (const float*)d_in[0];
  const float* B     = (const float*)d_in[1];
  const float* Q     = (const float*)d_in[2];
  const float* R     = (const float*)d_in[3];
  const float* goals = (const float*)d_in[4];
  float* out = (float*)d_out;
  koopman_lqr_kernel<<<dim3(BSZ), dim3(256), 0, stream>>>(A, B, Q, R, goals, out);
}